// RelGraphConvLayer_54116587930307
// MI455X (gfx1250) — compile-verified
//
#include <hip/hip_runtime.h>
#include <hip/hip_bf16.h>
#include <stdint.h>

#define N_NODES 100000
#define N_RELS  8
#define N_EDGES 150000
#define D       128    // D_IN == D_OUT
#define LDS_PAD 130    // padded K-stride for transposed W in LDS (floats)

typedef __attribute__((ext_vector_type(2))) float v2f;
typedef __attribute__((ext_vector_type(8))) float v8f;

// ---------------------------------------------------------------- zeroing ----
__global__ void zero4_kernel(float4* __restrict__ p, int n4) {
    int i = blockIdx.x * blockDim.x + threadIdx.x;
    int stride = gridDim.x * blockDim.x;
    float4 z = make_float4(0.f, 0.f, 0.f, 0.f);
    for (; i < n4; i += stride) p[i] = z;
}

// ---------------------------------------------------------------- scatter ----
// One wave per edge: 32 lanes x float4 = 128 floats. x is L2-resident (51MB),
// atomic fp32 adds land in L2 (msgs buffer also L2-resident).
__global__ void scatter_kernel(const float*   __restrict__ x,
                               const int64_t* __restrict__ src,
                               const int64_t* __restrict__ dst,
                               float*         __restrict__ msgs,
                               float*         __restrict__ deg,
                               int n_edges) {
    int gid  = blockIdx.x * blockDim.x + threadIdx.x;
    int edge = gid >> 5;
    int lane = gid & 31;
    if (edge >= n_edges) return;
    int64_t s = src[edge];
    int64_t d = dst[edge];
    const float4 v = *(const float4*)(x + (size_t)s * D + lane * 4);
    float* m = msgs + (size_t)d * D + lane * 4;
    atomicAdd(m + 0, v.x);
    atomicAdd(m + 1, v.y);
    atomicAdd(m + 2, v.z);
    atomicAdd(m + 3, v.w);
    if (lane == 0) atomicAdd(deg + d, 1.0f);
}

// ------------------------------------------------------------- WMMA GEMM -----
// out[N,128] = [out +] (A[N,128] * rowscale) @ W[128,128]  (fp32 WMMA 16x16x4)
// final_pass: also add bias and apply ReLU.
// Block = 256 threads = 8 waves; wave w owns row tile blockIdx.x*8 + w.
// W is staged TRANSPOSED in LDS (lwT[col][k], stride LDS_PAD) so each B
// fragment {W[k][col], W[k+1][col]} is one contiguous 64-bit DS load.
__global__ __launch_bounds__(256)
void gemm_acc_kernel(const float* __restrict__ A,
                     const float* __restrict__ deg,   // nullptr => scale 1
                     const float* __restrict__ W,
                     float*       __restrict__ out,
                     const float* __restrict__ bias,  // used when final_pass
                     int accumulate,
                     int final_pass) {
    __shared__ float lwT[D * LDS_PAD];   // 65 KB of the 320 KB WGP LDS
    {
        // Global: float4 along n (coalesced). LDS: 4 scalar transposed stores.
        const float4* w4 = (const float4*)W;
        for (int i = threadIdx.x; i < (D * D) / 4; i += 256) {
            int k  = i >> 5;          // row of W (K index)
            int n4 = i & 31;          // group of 4 columns
            float4 w = w4[i];
            float* t = lwT + (n4 * 4) * LDS_PAD + k;
            t[0 * LDS_PAD] = w.x;
            t[1 * LDS_PAD] = w.y;
            t[2 * LDS_PAD] = w.z;
            t[3 * LDS_PAD] = w.w;
        }
    }
    __syncthreads();

    const int wave    = threadIdx.x >> 5;
    const int lane    = threadIdx.x & 31;
    const int rowTile = blockIdx.x * 8 + wave;
    if (rowTile >= (N_NODES / 16)) return;   // wave-uniform: EXEC stays full

    const int row0 = rowTile * 16;
    const int m    = lane & 15;          // A row-in-tile / B,C column-in-tile
    const int kh   = (lane >> 4) * 2;    // K sub-offset within each K=4 chunk
    const int arow = row0 + m;

    float scale = 1.0f;
    if (deg) scale = 1.0f / fmaxf(deg[arow], 1.0f);

    // Preload A fragments for all 32 K-chunks, folding in the 1/deg scaling.
    v2f afrag[32];
    const float* arow_ptr = A + (size_t)arow * D + kh;
    #pragma unroll
    for (int kc = 0; kc < 32; ++kc) {
        v2f a = *(const v2f*)(arow_ptr + kc * 4);
        a[0] *= scale;
        a[1] *= scale;
        afrag[kc] = a;
    }

    v8f acc[8] = {};

    #pragma unroll 4
    for (int kc = 0; kc < 32; ++kc) {
        const v2f a  = afrag[kc];
        const int kk = kc * 4 + kh;
        const float* bcol = lwT + (size_t)m * LDS_PAD + kk;  // + nt*16*LDS_PAD
        #pragma unroll
        for (int nt = 0; nt < 8; ++nt) {
            const v2f b = *(const v2f*)(bcol + nt * 16 * LDS_PAD);
            acc[nt] = __builtin_amdgcn_wmma_f32_16x16x4_f32(
                false, a, false, b, (short)0, acc[nt], false, false);
        }
    }

    // C/D layout: VGPR v -> row row0 + v + (lane>=16 ? 8 : 0), col = lane&15.
    const int rbase = row0 + (lane >> 4) * 8;
    #pragma unroll
    for (int nt = 0; nt < 8; ++nt) {
        const int col = nt * 16 + m;
        const float bv = final_pass ? bias[col] : 0.0f;
        #pragma unroll
        for (int v = 0; v < 8; ++v) {
            const size_t idx = (size_t)(rbase + v) * D + col;
            float val = acc[nt][v];
            if (accumulate) val += out[idx];
            if (final_pass) val = fmaxf(val + bv, 0.0f);
            out[idx] = val;
        }
    }
}

// ------------------------------------------------------------------ launch ---
extern "C" void kernel_launch(void* const* d_in, const int* in_sizes, int n_in,
                              void* d_out, int out_size, void* d_ws, size_t ws_size,
                              hipStream_t stream) {
    (void)in_sizes; (void)n_in; (void)out_size; (void)ws_size;

    const float*   x      = (const float*)d_in[0];   // [N,128]
    const float*   weight = (const float*)d_in[1];   // [8,128,128]
    const float*   wself  = (const float*)d_in[2];   // [128,128]
    const float*   bias   = (const float*)d_in[3];   // [128]
    const int64_t* src    = (const int64_t*)d_in[4]; // [8,150000]
    const int64_t* dst    = (const int64_t*)d_in[5]; // [8,150000]

    float* out  = (float*)d_out;                       // [N,128]
    float* msgs = (float*)d_ws;                        // [N,128]
    float* deg  = msgs + (size_t)N_NODES * D;          // [N] (contiguous)

    const int n4_ws  = (N_NODES * D + N_NODES) / 4;    // msgs + deg together
    const int gemm_blocks    = (N_NODES / 16 + 7) / 8; // 782
    const int scatter_blocks = (N_EDGES * 32 + 255) / 256;

    for (int r = 0; r < N_RELS; ++r) {
        zero4_kernel<<<1024, 256, 0, stream>>>((float4*)msgs, n4_ws);
        scatter_kernel<<<scatter_blocks, 256, 0, stream>>>(
            x, src + (size_t)r * N_EDGES, dst + (size_t)r * N_EDGES,
            msgs, deg, N_EDGES);
        // First relation writes out directly (no RMW; out may hold poison).
        gemm_acc_kernel<<<gemm_blocks, 256, 0, stream>>>(
            msgs, deg, weight + (size_t)r * D * D, out, nullptr,
            /*accumulate=*/(r != 0), /*final_pass=*/0);
    }

    // Self-loop + bias + ReLU: A = x, scale = 1.
    gemm_acc_kernel<<<gemm_blocks, 256, 0, stream>>>(
        x, nullptr, wself, out, bias, /*accumulate=*/1, /*final_pass=*/1);
}